// AtomConvSum_80917183856993
// MI455X (gfx1250) — compile-verified
//
#include <hip/hip_runtime.h>
#include <math.h>

// AtomConvSum on gfx1250 (MI455X), fp32 throughout using V_WMMA_F32_16X16X4_F32.
// Bandwidth-bound problem: recompute core/gate in pass 2 instead of spilling
// 819MB of intermediates. Node-proj tables (102MB) + node accumulator (25.6MB)
// are L2-resident (192MB), so per-edge gathers and scatter atomics hit L2.

typedef __attribute__((ext_vector_type(2))) float v2f;
typedef __attribute__((ext_vector_type(8))) float v8f;

#define DIM 64

static __device__ __forceinline__ v8f wmma4(v2f a, v2f b, v8f c) {
    return __builtin_amdgcn_wmma_f32_16x16x4_f32(
        /*neg_a=*/false, a, /*neg_b=*/false, b,
        /*c_mod=*/(short)0, c, /*reuse_a=*/false, /*reuse_b=*/false);
}

__global__ void zero_kernel(float* __restrict__ p, int n) {
    int i = blockIdx.x * blockDim.x + threadIdx.x;
    if (i < n) p[i] = 0.0f;
}

// ---- node projections: P_w = X @ W_w^T for 4 weights, fused (X read once) ----
__global__ void node_proj_kernel(const float* __restrict__ X,
                                 const float* __restrict__ W0, const float* __restrict__ W1,
                                 const float* __restrict__ W2, const float* __restrict__ W3,
                                 float* __restrict__ P0, float* __restrict__ P1,
                                 float* __restrict__ P2, float* __restrict__ P3,
                                 int n_tiles) {
    const int lane = threadIdx.x & 31;
    const int wv   = threadIdx.x >> 5;
    const int tile = blockIdx.x * (blockDim.x >> 5) + wv;
    if (tile >= n_tiles) return;             // wave-uniform guard, EXEC stays full
    const int half = lane >> 4, l16 = lane & 15;
    const int rowA = tile * 16 + l16;

    const float* Ws[4] = {W0, W1, W2, W3};
    float*       Ps[4] = {P0, P1, P2, P3};

    v8f acc[4][4] = {};
    for (int k = 0; k < 16; ++k) {
        const int ko = 4 * k + 2 * half;     // K-pair owned by this half-wave
        v2f a = *(const v2f*)(X + rowA * DIM + ko);
        #pragma unroll
        for (int w = 0; w < 4; ++w) {
            #pragma unroll
            for (int nt = 0; nt < 4; ++nt) {
                v2f b = *(const v2f*)(Ws[w] + (nt * 16 + l16) * DIM + ko); // B[k][n]=W[n][k]
                acc[w][nt] = wmma4(a, b, acc[w][nt]);
            }
        }
    }
    #pragma unroll
    for (int w = 0; w < 4; ++w)
        #pragma unroll
        for (int nt = 0; nt < 4; ++nt)
            #pragma unroll
            for (int r = 0; r < 8; ++r)
                Ps[w][(tile * 16 + r + 8 * half) * DIM + nt * 16 + l16] = acc[w][nt][r];
}

// ---- shared edge-tile compute: core/gate = EF@Wb^T + gather(src)+gather(dst) ----
static __device__ __forceinline__ void edge_tile_compute(
    const float* __restrict__ EF,
    const float* __restrict__ Wcb, const float* __restrict__ Wgb,
    const float* __restrict__ Pcs, const float* __restrict__ Pcd,
    const float* __restrict__ Pgs, const float* __restrict__ Pgd,
    const int* __restrict__ srcI, const int* __restrict__ dstI,
    int tile, int lane, v8f core[4], v8f gate[4], int srcRows[8]) {

    const int half = lane >> 4, l16 = lane & 15;
    const int rowA = tile * 16 + l16;
    #pragma unroll
    for (int nt = 0; nt < 4; ++nt) { core[nt] = (v8f){}; gate[nt] = (v8f){}; }

    for (int k = 0; k < 16; ++k) {
        const int ko = 4 * k + 2 * half;
        v2f a = *(const v2f*)(EF + rowA * DIM + ko);
        #pragma unroll
        for (int nt = 0; nt < 4; ++nt) {
            v2f bc = *(const v2f*)(Wcb + (nt * 16 + l16) * DIM + ko);
            core[nt] = wmma4(a, bc, core[nt]);
            v2f bg = *(const v2f*)(Wgb + (nt * 16 + l16) * DIM + ko);
            gate[nt] = wmma4(a, bg, gate[nt]);
        }
    }
    // add gathered node projections; element (r,nt): row=edge, col=nt*16+l16
    #pragma unroll
    for (int r = 0; r < 8; ++r) {
        const int e = tile * 16 + r + 8 * half;
        const int s = srcI[e], d = dstI[e];
        srcRows[r] = s;
        #pragma unroll
        for (int nt = 0; nt < 4; ++nt) {
            const int col = nt * 16 + l16;
            core[nt][r] += Pcs[s * DIM + col] + Pcd[d * DIM + col];
            gate[nt][r] += Pgs[s * DIM + col] + Pgd[d * DIM + col];
        }
    }
}

// ---- pass 1: per-channel sum / sum-of-squares of core & gate ----
__global__ void edge_stats_kernel(const float* __restrict__ EF,
                                  const float* __restrict__ Wcb, const float* __restrict__ Wgb,
                                  const float* __restrict__ Pcs, const float* __restrict__ Pcd,
                                  const float* __restrict__ Pgs, const float* __restrict__ Pgd,
                                  const int* __restrict__ srcI, const int* __restrict__ dstI,
                                  float* __restrict__ sums, int n_tiles) {
    __shared__ float ls[256];
    for (int i = threadIdx.x; i < 256; i += blockDim.x) ls[i] = 0.0f;
    __syncthreads();

    const int lane = threadIdx.x & 31;
    const int wv   = threadIdx.x >> 5;
    const int wpb  = blockDim.x >> 5;
    const int gwave  = blockIdx.x * wpb + wv;
    const int nwaves = gridDim.x * wpb;

    float sc[4] = {}, sc2[4] = {}, sg[4] = {}, sg2[4] = {};
    for (int tile = gwave; tile < n_tiles; tile += nwaves) {
        v8f core[4], gate[4]; int srcRows[8];
        edge_tile_compute(EF, Wcb, Wgb, Pcs, Pcd, Pgs, Pgd, srcI, dstI,
                          tile, lane, core, gate, srcRows);
        #pragma unroll
        for (int nt = 0; nt < 4; ++nt)
            #pragma unroll
            for (int r = 0; r < 8; ++r) {
                float c = core[nt][r]; sc[nt] += c; sc2[nt] += c * c;
                float g = gate[nt][r]; sg[nt] += g; sg2[nt] += g * g;
            }
    }
    // lanes l and l+16 hold the same column -> pairwise reduce across halves
    #pragma unroll
    for (int nt = 0; nt < 4; ++nt) {
        sc[nt]  += __shfl_xor(sc[nt],  16, 32);
        sc2[nt] += __shfl_xor(sc2[nt], 16, 32);
        sg[nt]  += __shfl_xor(sg[nt],  16, 32);
        sg2[nt] += __shfl_xor(sg2[nt], 16, 32);
    }
    const int l16 = lane & 15;
    if (lane < 16) {
        #pragma unroll
        for (int nt = 0; nt < 4; ++nt) {
            const int col = nt * 16 + l16;
            atomicAdd(&ls[col],       sc[nt]);
            atomicAdd(&ls[64 + col],  sc2[nt]);
            atomicAdd(&ls[128 + col], sg[nt]);
            atomicAdd(&ls[192 + col], sg2[nt]);
        }
    }
    __syncthreads();
    for (int i = threadIdx.x; i < 256; i += blockDim.x) atomicAdd(&sums[i], ls[i]);
}

// ---- BN finalize: fold mean/var/gamma/beta into per-channel scale+shift ----
__global__ void bn_finalize_kernel(const float* __restrict__ sums,
                                   const float* __restrict__ gc, const float* __restrict__ bc,
                                   const float* __restrict__ gg, const float* __restrict__ bg,
                                   float* __restrict__ params, float inv_n) {
    const int i = threadIdx.x;
    if (i < DIM) {
        float mc = sums[i] * inv_n;
        float vc = sums[64 + i] * inv_n - mc * mc;
        float kc = gc[i] * rsqrtf(vc + 1e-5f);
        params[i]       = kc;
        params[64 + i]  = bc[i] - mc * kc;
        float mg = sums[128 + i] * inv_n;
        float vg = sums[192 + i] * inv_n - mg * mg;
        float kg = gg[i] * rsqrtf(vg + 1e-5f);
        params[128 + i] = kg;
        params[192 + i] = bg[i] - mg * kg;
    }
}

// ---- pass 2: recompute, normalize, silu*sigmoid, scatter-add into nodes ----
__global__ void edge_scatter_kernel(const float* __restrict__ EF,
                                    const float* __restrict__ Wcb, const float* __restrict__ Wgb,
                                    const float* __restrict__ Pcs, const float* __restrict__ Pcd,
                                    const float* __restrict__ Pgs, const float* __restrict__ Pgd,
                                    const int* __restrict__ srcI, const int* __restrict__ dstI,
                                    const float* __restrict__ params,
                                    float* __restrict__ accN, int n_tiles) {
    const int lane = threadIdx.x & 31;
    const int wv   = threadIdx.x >> 5;
    const int tile = blockIdx.x * (blockDim.x >> 5) + wv;
    if (tile >= n_tiles) return;
    const int l16 = lane & 15;

    v8f core[4], gate[4]; int srcRows[8];
    edge_tile_compute(EF, Wcb, Wgb, Pcs, Pcd, Pgs, Pgd, srcI, dstI,
                      tile, lane, core, gate, srcRows);

    float kc[4], hc[4], kg[4], hg[4];
    #pragma unroll
    for (int nt = 0; nt < 4; ++nt) {
        const int col = nt * 16 + l16;
        kc[nt] = params[col];       hc[nt] = params[64 + col];
        kg[nt] = params[128 + col]; hg[nt] = params[192 + col];
    }
    #pragma unroll
    for (int r = 0; r < 8; ++r) {
        const int s = srcRows[r];
        #pragma unroll
        for (int nt = 0; nt < 4; ++nt) {
            const int col = nt * 16 + l16;
            float cn = core[nt][r] * kc[nt] + hc[nt];
            float gn = gate[nt][r] * kg[nt] + hg[nt];
            float silu = cn / (1.0f + __expf(-cn));
            float sigg = 1.0f / (1.0f + __expf(-gn));
            atomicAdd(&accN[s * DIM + col], silu * sigg);
        }
    }
}

// ---- output: OUT = ACC @ Wout^T + X ----
__global__ void out_proj_kernel(const float* __restrict__ ACC,
                                const float* __restrict__ Wout,
                                const float* __restrict__ X,
                                float* __restrict__ OUT, int n_tiles) {
    const int lane = threadIdx.x & 31;
    const int wv   = threadIdx.x >> 5;
    const int tile = blockIdx.x * (blockDim.x >> 5) + wv;
    if (tile >= n_tiles) return;
    const int half = lane >> 4, l16 = lane & 15;
    const int rowA = tile * 16 + l16;

    v8f acc[4] = {};
    for (int k = 0; k < 16; ++k) {
        const int ko = 4 * k + 2 * half;
        v2f a = *(const v2f*)(ACC + rowA * DIM + ko);
        #pragma unroll
        for (int nt = 0; nt < 4; ++nt) {
            v2f b = *(const v2f*)(Wout + (nt * 16 + l16) * DIM + ko);
            acc[nt] = wmma4(a, b, acc[nt]);
        }
    }
    #pragma unroll
    for (int nt = 0; nt < 4; ++nt)
        #pragma unroll
        for (int r = 0; r < 8; ++r) {
            const int row = tile * 16 + r + 8 * half;
            const int col = nt * 16 + l16;
            OUT[row * DIM + col] = acc[nt][r] + X[row * DIM + col];
        }
}

extern "C" void kernel_launch(void* const* d_in, const int* in_sizes, int n_in,
                              void* d_out, int out_size, void* d_ws, size_t ws_size,
                              hipStream_t stream) {
    const float* X   = (const float*)d_in[0];
    const float* EF  = (const float*)d_in[1];
    const int*   EI  = (const int*)d_in[2];
    const float* Wcs = (const float*)d_in[3];
    const float* Wcd = (const float*)d_in[4];
    const float* Wcb = (const float*)d_in[5];
    const float* Wgs = (const float*)d_in[6];
    const float* Wgd = (const float*)d_in[7];
    const float* Wgb = (const float*)d_in[8];
    const float* gc  = (const float*)d_in[9];
    const float* bc  = (const float*)d_in[10];
    const float* gg  = (const float*)d_in[11];
    const float* bg  = (const float*)d_in[12];
    const float* Wout = (const float*)d_in[13];
    float* out = (float*)d_out;

    const int n_nodes = in_sizes[0] / DIM;
    const int n_edges = in_sizes[1] / DIM;
    const int* srcI = EI;
    const int* dstI = EI + n_edges;

    // workspace layout (floats): Pcs|Pcd|Pgs|Pgd|accN|sums(256)|params(256)
    float* ws = (float*)d_ws;
    const size_t np = (size_t)n_nodes * DIM;
    float* Pcs    = ws;
    float* Pcd    = Pcs + np;
    float* Pgs    = Pcd + np;
    float* Pgd    = Pgs + np;
    float* accN   = Pgd + np;
    float* sums   = accN + np;
    float* params = sums + 256;

    const int node_tiles = (n_nodes + 15) / 16;   // 6250
    const int edge_tiles = (n_edges + 15) / 16;   // 100000

    const int zc = (int)np + 256;                 // zero accN + sums each call
    zero_kernel<<<(zc + 255) / 256, 256, 0, stream>>>(accN, zc);

    node_proj_kernel<<<(node_tiles + 7) / 8, 256, 0, stream>>>(
        X, Wcs, Wcd, Wgs, Wgd, Pcs, Pcd, Pgs, Pgd, node_tiles);

    edge_stats_kernel<<<1024, 256, 0, stream>>>(
        EF, Wcb, Wgb, Pcs, Pcd, Pgs, Pgd, srcI, dstI, sums, edge_tiles);

    bn_finalize_kernel<<<1, 64, 0, stream>>>(
        sums, gc, bc, gg, bg, params, 1.0f / (float)n_edges);

    edge_scatter_kernel<<<(edge_tiles + 7) / 8, 256, 0, stream>>>(
        EF, Wcb, Wgb, Pcs, Pcd, Pgs, Pgd, srcI, dstI, params, accN, edge_tiles);

    out_proj_kernel<<<(node_tiles + 7) / 8, 256, 0, stream>>>(
        accN, Wout, X, out, node_tiles);
}